// VertexNormals_32091995636360
// MI455X (gfx1250) — compile-verified
//
#include <hip/hip_runtime.h>

// Vertex-normal computation for B=32, V=200000, F=400000 (float32).
// zero -> face-normal atomic scatter (async-LDS staged face stream)
//      -> in-place normalize.
// Whole working set (~160 MB) lives in MI455X's 192 MB L2; bottleneck is the
// ~115M non-returning global_atomic_add_f32 ops handled by L2 atomic units.

#define EPSF 1e-12f
static constexpr int kV   = 200000;  // vertices per batch (fixed by reference)
static constexpr int TILE = 256;     // faces per tile == blockDim.x

// ---- CDNA5 async global->LDS staging (guarded; falls back to plain LDS) ----
#if __has_builtin(__builtin_amdgcn_global_load_async_to_lds_b32)
#define HAVE_ASYNC_LDS 1
typedef __attribute__((address_space(1))) int* as1i;
typedef __attribute__((address_space(3))) int* as3i;
__device__ __forceinline__ void async_g2l_b32(const int* g, int* l) {
    __builtin_amdgcn_global_load_async_to_lds_b32(
        (as1i)(uintptr_t)g, (as3i)(uint32_t)(uintptr_t)l, /*offset=*/0, /*cpol=*/0);
}
__device__ __forceinline__ void wait_async0() {
#if __has_builtin(__builtin_amdgcn_s_wait_asynccnt)
    __builtin_amdgcn_s_wait_asynccnt(0);
#else
    asm volatile("s_wait_asynccnt 0x0" ::: "memory");
#endif
}
#else
#define HAVE_ASYNC_LDS 0
#endif

// ---------------------------------------------------------------- zero ----
__global__ void vn_zero_kernel(float* __restrict__ p, long n) {
    const long n4 = n >> 2;
    float4* p4 = (float4*)p;
    long i = (long)blockIdx.x * blockDim.x + threadIdx.x;
    const long stride = (long)gridDim.x * blockDim.x;
    const float4 z = make_float4(0.f, 0.f, 0.f, 0.f);
    for (; i < n4; i += stride) p4[i] = z;
    if (blockIdx.x == 0 && threadIdx.x == 0) {
        for (long k = n4 << 2; k < n; ++k) p[k] = 0.f;
    }
}

// ------------------------------------------------------------- scatter ----
// grid.y = batch; blocks grid-stride over 256-face tiles.
// Face indices are staged coalesced into LDS via async copy, vertex gathers
// hit the L2-resident per-batch slab, normals scatter via 9 f32 atomics.
__global__ void vn_scatter_kernel(const float* __restrict__ verts,
                                  const int*   __restrict__ faces,
                                  float*       __restrict__ out,
                                  int F, int V) {
    __shared__ int tile[3 * TILE];

    const int b = blockIdx.y;
    const float* vb = verts + (long)b * V * 3;
    float*       ob = out   + (long)b * V * 3;
    const int t = threadIdx.x;

    for (int tileBase = blockIdx.x * TILE; tileBase < F;
         tileBase += gridDim.x * TILE) {
        const int nFaces = min(TILE, F - tileBase);
        const int nInts  = 3 * nFaces;
        const int* gsrc  = faces + 3L * tileBase;

#if HAVE_ASYNC_LDS
        for (int k = t; k < nInts; k += TILE)          // ASYNCcnt, no VGPR trip
            async_g2l_b32(gsrc + k, &tile[k]);
        wait_async0();
#else
        for (int k = t; k < nInts; k += TILE)          // plain coalesced stage
            tile[k] = gsrc[k];
#endif
        __syncthreads();

        if (t < nFaces) {
            const int i0 = tile[3 * t + 0];
            const int i1 = tile[3 * t + 1];
            const int i2 = tile[3 * t + 2];

            const long o0 = 3L * i0, o1 = 3L * i1, o2 = 3L * i2;
            const float v0x = vb[o0 + 0], v0y = vb[o0 + 1], v0z = vb[o0 + 2];
            const float v1x = vb[o1 + 0], v1y = vb[o1 + 1], v1z = vb[o1 + 2];
            const float v2x = vb[o2 + 0], v2y = vb[o2 + 1], v2z = vb[o2 + 2];

            // e1 = v0 - v1 ; e2 = v2 - v1 ; n = cross(e2, e1)
            const float e1x = v0x - v1x, e1y = v0y - v1y, e1z = v0z - v1z;
            const float e2x = v2x - v1x, e2y = v2y - v1y, e2z = v2z - v1z;
            const float nx = e2y * e1z - e2z * e1y;
            const float ny = e2z * e1x - e2x * e1z;
            const float nz = e2x * e1y - e2y * e1x;

            // 9 fire-and-forget f32 atomics -> L2 atomic units (STOREcnt).
            unsafeAtomicAdd(&ob[o0 + 0], nx);
            unsafeAtomicAdd(&ob[o0 + 1], ny);
            unsafeAtomicAdd(&ob[o0 + 2], nz);
            unsafeAtomicAdd(&ob[o1 + 0], nx);
            unsafeAtomicAdd(&ob[o1 + 1], ny);
            unsafeAtomicAdd(&ob[o1 + 2], nz);
            unsafeAtomicAdd(&ob[o2 + 0], nx);
            unsafeAtomicAdd(&ob[o2 + 1], ny);
            unsafeAtomicAdd(&ob[o2 + 2], nz);
        }
        __syncthreads();   // protect tile[] before next stage
    }
}

// ----------------------------------------------------------- normalize ----
__global__ void vn_normalize_kernel(float* __restrict__ out, long nVert) {
    long i = (long)blockIdx.x * blockDim.x + threadIdx.x;
    const long stride = (long)gridDim.x * blockDim.x;
    for (; i < nVert; i += stride) {
        const long base = 3L * i;
        const float x = out[base + 0];
        const float y = out[base + 1];
        const float z = out[base + 2];
        const float m = fmaxf(fmaf(x, x, fmaf(y, y, z * z)), EPSF);
        float s = rsqrtf(m);
        s = s * (1.5f - 0.5f * m * s * s);   // one Newton step -> ~full f32
        out[base + 0] = x * s;
        out[base + 1] = y * s;
        out[base + 2] = z * s;
    }
}

// --------------------------------------------------------------- launch ---
extern "C" void kernel_launch(void* const* d_in, const int* in_sizes, int n_in,
                              void* d_out, int out_size, void* d_ws, size_t ws_size,
                              hipStream_t stream) {
    const float* verts = (const float*)d_in[0];   // (B, V, 3) f32
    const int*   faces = (const int*)d_in[1];     // (F, 3)    i32
    float*       out   = (float*)d_out;           // (B, V, 3) f32

    const int V = kV;
    const int F = in_sizes[1] / 3;
    const int B = (in_sizes[0] / 3) / V;

    const long nTotal = (long)out_size;           // B*V*3 floats
    const long nVert  = (long)B * V;

    const int tb = TILE;                          // 8 wave32 waves / block

    // 1) zero accumulator
    {
        long n4 = nTotal >> 2;
        int blocks = (int)((n4 + tb - 1) / tb);
        vn_zero_kernel<<<blocks, tb, 0, stream>>>(out, nTotal);
    }

    // 2) face-normal scatter (grid.y = batch)
    {
        dim3 grid(512, (unsigned)B, 1);
        vn_scatter_kernel<<<grid, tb, 0, stream>>>(verts, faces, out, F, V);
    }

    // 3) normalize in place
    {
        int blocks = (int)((nVert + tb - 1) / tb);
        vn_normalize_kernel<<<blocks, tb, 0, stream>>>(out, nVert);
    }
}